// HadamardExpansionV2_3968549781674
// MI455X (gfx1250) — compile-verified
//
#include <hip/hip_runtime.h>
#include <stdint.h>

typedef float floatx4 __attribute__((ext_vector_type(4)));
typedef int   intx4   __attribute__((__vector_size__(16)));
typedef __attribute__((address_space(1))) intx4 g_intx4;
typedef __attribute__((address_space(3))) intx4 l_intx4;

#define C1V 256          // input channels
#define CEV 512          // expanded channels
#define BV  16           // batch
#define HW4 784          // (56*56)/4 float4 per (b, channel) slice
#define STAGE_ASYNC_OPS 8

// ---- CDNA5 async global->LDS path (guarded; falls back to plain loads) ----
#if defined(__has_builtin)
# if __has_builtin(__builtin_amdgcn_global_load_async_to_lds_b128)
#  define USE_ASYNC_LDS 1
# endif
#endif

#ifdef USE_ASYNC_LDS
# if __has_builtin(__builtin_amdgcn_s_wait_asynccnt)
#  define WAIT_ASYNC(n) __builtin_amdgcn_s_wait_asynccnt(n)
# else
#  define WAIT_ASYNC(n) asm volatile("s_wait_asynccnt " #n)
# endif
#else
# define WAIT_ASYNC(n) ((void)0)
#endif

__device__ __forceinline__ void async_copy_f4(const floatx4* g, floatx4* l) {
#ifdef USE_ASYNC_LDS
  __builtin_amdgcn_global_load_async_to_lds_b128((g_intx4*)g, (l_intx4*)l, 0, 0);
#else
  *l = *g;
#endif
}

// Stage one (b, c0) and one (b, c1) slice into LDS tiles.
// Each thread writes ONLY its own slots (tail lanes re-stage slot `tid`,
// keeping the per-wave async count uniform at 8) -> no cross-thread LDS
// hazards, so per-wave s_wait_asynccnt is the only sync needed.
__device__ __forceinline__ void stage_pair(const floatx4* __restrict__ g0,
                                           const floatx4* __restrict__ g1,
                                           floatx4* t0, floatx4* t1, int tid) {
#pragma unroll
  for (int k = 0; k < 4; ++k) {
    int i = tid + (k << 8);
    int j = (i < HW4) ? i : tid;   // owner-only duplicate for uniform count
    async_copy_f4(g0 + j, t0 + j);
    async_copy_f4(g1 + j, t1 + j);
  }
}

// One wave computes argmax over a 256-element row of logits+gumbel.
// First-occurrence tie-break (matches jnp.argmax); all lanes return winner.
__device__ __forceinline__ int wave_argmax_row(const floatx4* __restrict__ L4,
                                               const floatx4* __restrict__ G4,
                                               int lane) {
  float bv = -__builtin_inff();
  int bi = 0;
#pragma unroll
  for (int k = 0; k < 2; ++k) {
    int q = lane + (k << 5);                    // float4 index 0..63
    floatx4 s = L4[q] + G4[q];
    int c = q << 2;
    if (s.x > bv || (s.x == bv && c     < bi)) { bv = s.x; bi = c;     }
    if (s.y > bv || (s.y == bv && c + 1 < bi)) { bv = s.y; bi = c + 1; }
    if (s.z > bv || (s.z == bv && c + 2 < bi)) { bv = s.z; bi = c + 2; }
    if (s.w > bv || (s.w == bv && c + 3 < bi)) { bv = s.w; bi = c + 3; }
  }
#pragma unroll
  for (int off = 16; off > 0; off >>= 1) {
    float ov = __shfl_xor(bv, off, 32);
    int   oi = __shfl_xor(bi, off, 32);
    if (ov > bv || (ov == bv && oi < bi)) { bv = ov; bi = oi; }
  }
  return bi;
}

// Fused: per-block argmax pair -> gather channel pair -> multiply -> BN.
// One block per expanded channel e. Two passes over (B, H*W); x stays in L2.
__global__ __launch_bounds__(256) void hadamard_expand_bn_kernel(
    const float* __restrict__ xf, const float* __restrict__ logits,
    const float* __restrict__ gumbel, const float* __restrict__ gamma,
    const float* __restrict__ beta, float* __restrict__ outf) {
  __shared__ floatx4 tile0[2][HW4];   // double-buffered c0 slice
  __shared__ floatx4 tile1[2][HW4];   // double-buffered c1 slice
  __shared__ float red0[256];
  __shared__ float red1[256];
  __shared__ float s_scale, s_shift;
  __shared__ int   s_c01[2];

  const int e   = blockIdx.x;
  const int tid = threadIdx.x;
  const int lane = tid & 31;

  // -------- Fused gumbel-softmax hard selection (waves 0 and 1 only) -------
  // mask = one_hot(argmax(logits + gumbel)); softmax & tau>0 are monotone.
  if (tid < 64) {
    const int q = tid >> 5;                       // 0 -> mask[0], 1 -> mask[1]
    const size_t r = (size_t)(q * CEV + e) * 64;  // row base in float4s
    int bi = wave_argmax_row((const floatx4*)logits + r,
                             (const floatx4*)gumbel + r, lane);
    if (lane == 0) s_c01[q] = bi;
  }
  __syncthreads();
  const int c0 = s_c01[0];
  const int c1 = s_c01[1];

  const floatx4* x4 = (const floatx4*)xf;
  floatx4* out4 = (floatx4*)outf;

  // ---------------- Pass 1: per-channel sum / sumsq ----------------
  float sum = 0.f, ssq = 0.f;
  stage_pair(x4 + (size_t)c0 * HW4, x4 + (size_t)c1 * HW4,
             tile0[0], tile1[0], tid);
  for (int b = 0; b < BV; ++b) {
    if (b + 1 < BV) {
      stage_pair(x4 + (size_t)((b + 1) * C1V + c0) * HW4,
                 x4 + (size_t)((b + 1) * C1V + c1) * HW4,
                 tile0[(b + 1) & 1], tile1[(b + 1) & 1], tid);
      WAIT_ASYNC(STAGE_ASYNC_OPS);   // stage b done; stage b+1 in flight
    } else {
      WAIT_ASYNC(0);
    }
    const floatx4* t0 = tile0[b & 1];
    const floatx4* t1 = tile1[b & 1];
#pragma unroll
    for (int k = 0; k < 4; ++k) {
      int i = tid + (k << 8);
      if (i < HW4) {
        floatx4 p = t0[i] * t1[i];
        sum += (p.x + p.y) + (p.z + p.w);
        ssq += (p.x * p.x + p.y * p.y) + (p.z * p.z + p.w * p.w);
      }
    }
  }

  // ---------------- Block reduction -> mean, rstd ----------------
  red0[tid] = sum;
  red1[tid] = ssq;
  __syncthreads();
#pragma unroll
  for (int s = 128; s > 0; s >>= 1) {
    if (tid < s) { red0[tid] += red0[tid + s]; red1[tid] += red1[tid + s]; }
    __syncthreads();
  }
  if (tid == 0) {
    const float inv  = 1.0f / (float)(BV * HW4 * 4);   // 1/50176
    float mean = red0[0] * inv;
    float var  = red1[0] * inv - mean * mean;
    float rstd = rsqrtf(var + 1e-5f);
    float sc   = gamma[e] * rstd;
    s_scale = sc;
    s_shift = beta[e] - mean * sc;
  }
  __syncthreads();
  const float scale = s_scale;
  const float shift = s_shift;

  // ---------------- Pass 2: normalize + NT store ----------------
  stage_pair(x4 + (size_t)c0 * HW4, x4 + (size_t)c1 * HW4,
             tile0[0], tile1[0], tid);
  for (int b = 0; b < BV; ++b) {
    if (b + 1 < BV) {
      stage_pair(x4 + (size_t)((b + 1) * C1V + c0) * HW4,
                 x4 + (size_t)((b + 1) * C1V + c1) * HW4,
                 tile0[(b + 1) & 1], tile1[(b + 1) & 1], tid);
      WAIT_ASYNC(STAGE_ASYNC_OPS);
    } else {
      WAIT_ASYNC(0);
    }
    const floatx4* t0 = tile0[b & 1];
    const floatx4* t1 = tile1[b & 1];
    floatx4* dst = out4 + (size_t)(b * CEV + e) * HW4;
#pragma unroll
    for (int k = 0; k < 4; ++k) {
      int i = tid + (k << 8);
      if (i < HW4) {
        floatx4 p = t0[i] * t1[i];
        floatx4 o = p * scale + shift;
        __builtin_nontemporal_store(o, dst + i);   // write-once output: NT
      }
    }
  }
}

extern "C" void kernel_launch(void* const* d_in, const int* in_sizes, int n_in,
                              void* d_out, int out_size, void* d_ws, size_t ws_size,
                              hipStream_t stream) {
  const float* x      = (const float*)d_in[0];  // [16,256,56,56]
  const float* logits = (const float*)d_in[1];  // [2,512,256]
  const float* gumbel = (const float*)d_in[2];  // [2,512,256]
  // d_in[3] = tau: positive scalar, argmax-invariant -> unused in forward
  const float* gamma  = (const float*)d_in[4];  // [512]
  const float* beta   = (const float*)d_in[5];  // [512]
  (void)d_ws; (void)ws_size; (void)in_sizes; (void)n_in; (void)out_size;

  hadamard_expand_bn_kernel<<<CEV, 256, 0, stream>>>(x, logits, gumbel,
                                                     gamma, beta,
                                                     (float*)d_out);
}